// GAT_3040836846450
// MI455X (gfx1250) — compile-verified
//
#include <hip/hip_runtime.h>
#include <hip/hip_bf16.h>

#define D_DIM   128
#define N_NODES 50001
#define N_EDGES 800000

typedef float v2f __attribute__((ext_vector_type(2)));
typedef float v8f __attribute__((ext_vector_type(8)));

// ---------------------------------------------------------------------------
// Kernel 1: item_scaled[n][f] = sum_d emb[n][d] * W[d][f] + b[f]
// One wave (32 lanes) per 16-row M strip, full N=128 via 8 accumulator tiles.
// Uses V_WMMA_F32_16X16X4_F32 (fp32 matrix core path, no precision loss).
// A 16x4 layout (ISA 7.12.2): lane l holds M=l&15, K = k + 2*(l>>4) + {0,1}.
// B 4x16 assumed symmetric: lane l holds N=l&15, K = k + 2*(l>>4) + {0,1}.
// C/D 16x16: VGPR r, lane l -> M = r + 8*(l>>4), N = l&15.
// Out-of-range A rows are address-clamped (not zeroed): garbage only lands in
// D rows that the epilogue never stores, so EXEC stays all-1s with no
// per-lane selects or exec-mask juggling in the hot loop.
// ---------------------------------------------------------------------------
__global__ __launch_bounds__(32) void k_scale_wmma(const float* __restrict__ emb,
                                                   const float* __restrict__ W,
                                                   const float* __restrict__ bsc,
                                                   float* __restrict__ item) {
    const int lane = threadIdx.x & 31;
    const int half = lane >> 4;          // 0 or 1
    const int l15  = lane & 15;
    const int mbase = blockIdx.x * 16;

    const int row  = mbase + l15;                          // A row this lane feeds
    const int rowc = row < N_NODES ? row : (N_NODES - 1);  // clamp address only

    v8f acc[8];
#pragma unroll
    for (int nt = 0; nt < 8; ++nt) {
        const float bv = bsc[nt * 16 + l15];               // bias depends on N only
#pragma unroll
        for (int r = 0; r < 8; ++r) acc[nt][r] = bv;
    }

    const float* arow = emb + (size_t)rowc * D_DIM;

#pragma unroll 1
    for (int k = 0; k < D_DIM; k += 4) {
        const int kk = k + 2 * half;
        const v2f a = *(const v2f*)(arow + kk);            // one global_load_b64
#pragma unroll
        for (int nt = 0; nt < 8; ++nt) {
            const int n = nt * 16 + l15;
            v2f b;
            b.x = W[(size_t)kk * D_DIM + n];
            b.y = W[(size_t)(kk + 1) * D_DIM + n];
            acc[nt] = __builtin_amdgcn_wmma_f32_16x16x4_f32(
                false, a, false, b, (short)0, acc[nt], false, false);
        }
    }

    if (mbase + 16 <= N_NODES) {
        // wave-uniform fast path: whole tile valid (3125 of 3126 blocks)
#pragma unroll
        for (int nt = 0; nt < 8; ++nt) {
#pragma unroll
            for (int r = 0; r < 8; ++r) {
                const int m = mbase + r + 8 * half;
                item[(size_t)m * D_DIM + nt * 16 + l15] = acc[nt][r];
            }
        }
    } else {
        // tail tile: per-row guard
#pragma unroll
        for (int nt = 0; nt < 8; ++nt) {
#pragma unroll
            for (int r = 0; r < 8; ++r) {
                const int m = mbase + r + 8 * half;
                if (m < N_NODES) {
                    item[(size_t)m * D_DIM + nt * 16 + l15] = acc[nt][r];
                }
            }
        }
    }
}

// ---------------------------------------------------------------------------
// Kernel 2: per-node attention dots.
//   a_src[n] = item[n] . W_att[0:128]
//   a_dst[n] = item[n] . W_att[128:256]
// One wave per node: 32 lanes x float4 = 128 elements, shuffle reduction.
// ---------------------------------------------------------------------------
__global__ __launch_bounds__(256) void k_att(const float* __restrict__ item,
                                             const float* __restrict__ Watt,
                                             float* __restrict__ asrc,
                                             float* __restrict__ adst) {
    const int wave = threadIdx.x >> 5;
    const int lane = threadIdx.x & 31;
    const int n = blockIdx.x * 8 + wave;
    if (n >= N_NODES) return;

    const float4 x  = ((const float4*)(item + (size_t)n * D_DIM))[lane];
    const float4 wa = ((const float4*)Watt)[lane];
    const float4 wb = ((const float4*)(Watt + D_DIM))[lane];

    float s1 = x.x * wa.x + x.y * wa.y + x.z * wa.z + x.w * wa.w;
    float s2 = x.x * wb.x + x.y * wb.y + x.z * wb.z + x.w * wb.w;
#pragma unroll
    for (int off = 16; off > 0; off >>= 1) {
        s1 += __shfl_down(s1, off, 32);
        s2 += __shfl_down(s2, off, 32);
    }
    if (lane == 0) {
        asrc[n] = s1;
        adst[n] = s2;
    }
}

// ---------------------------------------------------------------------------
// Kernel 3: segmented aggregation + sigmoid. One wave per src node.
// src column of the (sorted-by-src) int64 edge array is binary-searched for
// the segment; scores, normalization, and the weighted gather-sum are fused:
//   out[n] = sigmoid( (sum_e sc[e] * item[dst[e]]) / sum_e sc[e] )
// Empty segments -> sigmoid(0) = 0.5 (matches reference).
// ---------------------------------------------------------------------------
__global__ __launch_bounds__(256) void k_agg(const long long* __restrict__ edge,
                                             const float* __restrict__ item,
                                             const float* __restrict__ asrc,
                                             const float* __restrict__ adst,
                                             const float* __restrict__ batt,
                                             float* __restrict__ out) {
    const int wave = threadIdx.x >> 5;
    const int lane = threadIdx.x & 31;
    const int n = blockIdx.x * 8 + wave;
    if (n >= N_NODES) return;

    // lower_bound(n) over edge[2e] -- wave-uniform search
    int lo = 0, hi = N_EDGES;
    while (lo < hi) {
        const int mid = (lo + hi) >> 1;
        if ((int)edge[2 * (size_t)mid] < n) lo = mid + 1; else hi = mid;
    }
    const int beg = lo;
    hi = N_EDGES;
    while (lo < hi) {
        const int mid = (lo + hi) >> 1;
        if ((int)edge[2 * (size_t)mid] < n + 1) lo = mid + 1; else hi = mid;
    }
    const int end = lo;

    const float as = asrc[n];
    const float bb = batt[0];

    float4 acc = make_float4(0.f, 0.f, 0.f, 0.f);
    float  ssum = 0.f;
    for (int e = beg; e < end; ++e) {
        const int d = (int)edge[2 * (size_t)e + 1];
        float att = as + adst[d] + bb;
        att = (att >= 0.f) ? att : 0.2f * att;             // LeakyReLU(0.2)
        const float sc = __expf(att - 1.0f);
        ssum += sc;
        const float4 x = ((const float4*)(item + (size_t)d * D_DIM))[lane];
        acc.x += sc * x.x;
        acc.y += sc * x.y;
        acc.z += sc * x.z;
        acc.w += sc * x.w;
    }

    const float inv = (end > beg) ? (1.0f / ssum) : 0.0f;
    float4 o;
    o.x = 1.0f / (1.0f + __expf(-acc.x * inv));
    o.y = 1.0f / (1.0f + __expf(-acc.y * inv));
    o.z = 1.0f / (1.0f + __expf(-acc.z * inv));
    o.w = 1.0f / (1.0f + __expf(-acc.w * inv));
    ((float4*)(out + (size_t)n * D_DIM))[lane] = o;
}

// ---------------------------------------------------------------------------
extern "C" void kernel_launch(void* const* d_in, const int* in_sizes, int n_in,
                              void* d_out, int out_size, void* d_ws, size_t ws_size,
                              hipStream_t stream) {
    const long long* edge = (const long long*)d_in[0];   // (E,2) int64
    const float* emb      = (const float*)d_in[1];       // (N_NODES,128) f32
    const float* Wsc      = (const float*)d_in[2];       // (128,128)
    const float* bsc      = (const float*)d_in[3];       // (128,)
    const float* Watt     = (const float*)d_in[4];       // (256,)
    const float* batt     = (const float*)d_in[5];       // (1,)
    float* out            = (float*)d_out;               // (N_NODES,128)

    // Workspace layout (every byte written before read; no zero-init needed)
    char* ws = (char*)d_ws;
    float* item = (float*)ws;                                   // N_NODES*128 f32
    size_t item_bytes = ((size_t)N_NODES * D_DIM * sizeof(float) + 255) & ~(size_t)255;
    float* asrc = (float*)(ws + item_bytes);                    // N_NODES f32
    size_t a_bytes = ((size_t)N_NODES * sizeof(float) + 255) & ~(size_t)255;
    float* adst = (float*)(ws + item_bytes + a_bytes);          // N_NODES f32

    const int m_tiles    = (N_NODES + 15) / 16;   // 3126
    const int node_blks  = (N_NODES + 7) / 8;     // 6251

    k_scale_wmma<<<m_tiles, 32, 0, stream>>>(emb, Wsc, bsc, item);
    k_att<<<node_blks, 256, 0, stream>>>(item, Watt, asrc, adst);
    k_agg<<<node_blks, 256, 0, stream>>>(edge, item, asrc, adst, batt, out);
}